// GCN_8392366096425
// MI455X (gfx1250) — compile-verified
//
#include <hip/hip_runtime.h>

typedef float v2f __attribute__((ext_vector_type(2)));
typedef float v8f __attribute__((ext_vector_type(8)));

#define TPB 256

// ---------------- utility kernels ----------------

__global__ void zero_f32(float* __restrict__ p, long long n) {
    long long i = (long long)blockIdx.x * blockDim.x + threadIdx.x;
    if (i < n) p[i] = 0.0f;
}

__global__ void degree_count(const int* __restrict__ src, const int* __restrict__ dst,
                             float* __restrict__ degOut, float* __restrict__ degIn, int nE) {
    int e = blockIdx.x * blockDim.x + threadIdx.x;
    if (e < nE) {
        atomicAdd(&degOut[src[e]], 1.0f);
        atomicAdd(&degIn[dst[e]], 1.0f);
    }
}

__global__ void degree_finalize(float* __restrict__ deg, long long n) {
    long long i = (long long)blockIdx.x * blockDim.x + threadIdx.x;
    if (i < n) deg[i] = rsqrtf(fmaxf(deg[i], 1.0f));
}

// ---------------- WMMA GEMM with fused row scaling ----------------
// Y[m, colBase+n] = rowScale[m] * sum_k X[m,k] * W[k, colBase+n]
// K fixed at 128. Block = 256 threads = 8 waves; each wave -> one 16x16 tile,
// block covers 128 rows x 16 cols. X staged through LDS in two K-chunks of 64.

template<int FOUT>
__global__ __launch_bounds__(256)
void gemm_rowscale(const float* __restrict__ X, const float* __restrict__ W,
                   const float* __restrict__ rowScale, float* __restrict__ Y,
                   int nRows)
{
    constexpr int K = 128;
    __shared__ float ldsX[128][65];   // [local row][k within 64-chunk], pad -> odd stride
    __shared__ float ldsW[K][16];     // [k][local col]

    const int t    = threadIdx.x;
    const int lane = t & 31;
    const int wave = t >> 5;
    const int hi   = lane >> 4;       // half-wave select (0: lanes 0-15, 1: lanes 16-31)
    const int mn   = lane & 15;       // M (for A) / N (for B,C,D) within tile
    const int rowBase = blockIdx.x * 128;
    const int colBase = blockIdx.y * 16;

    // Stage W tile once: 128 x 16 elements, 8 per thread.
#pragma unroll
    for (int i = 0; i < 8; ++i) {
        int idx = t + TPB * i;            // 0..2047
        int k = idx >> 4, c = idx & 15;
        ldsW[k][c] = W[(size_t)k * FOUT + colBase + c];
    }

    const int rloc = wave * 16 + mn;      // local row this lane reads for A
    v8f acc = {};

    // Staging split for X: thread t loads 32 floats of row (t>>1), col half (t&1)*32
    const int srow  = t >> 1;
    const int scol0 = (t & 1) * 32;
    const int grow  = rowBase + srow;
    const bool rowOK = (grow < nRows);

    for (int kc = 0; kc < 2; ++kc) {
        __syncthreads();                  // protect ldsX reuse across chunks (also covers ldsW on kc==0)
        const int kbase = kc * 64;
#pragma unroll
        for (int i = 0; i < 8; ++i) {
            float4 v = make_float4(0.f, 0.f, 0.f, 0.f);
            if (rowOK)
                v = *(const float4*)(X + (size_t)grow * K + kbase + scol0 + 4 * i);
            ldsX[srow][scol0 + 4 * i + 0] = v.x;
            ldsX[srow][scol0 + 4 * i + 1] = v.y;
            ldsX[srow][scol0 + 4 * i + 2] = v.z;
            ldsX[srow][scol0 + 4 * i + 3] = v.w;
        }
        __syncthreads();

#pragma unroll
        for (int k0 = 0; k0 < 64; k0 += 4) {
            // A 16x4: lane holds M=mn; vgpr0 = K(k0 + hi*2), vgpr1 = K(k0 + hi*2 + 1)
            v2f a, b;
            a.x = ldsX[rloc][k0 + hi * 2 + 0];
            a.y = ldsX[rloc][k0 + hi * 2 + 1];
            // B 4x16: lane holds N=mn; vgpr0 = row K(k0 + hi*2), vgpr1 = next K row
            b.x = ldsW[kbase + k0 + hi * 2 + 0][mn];
            b.y = ldsW[kbase + k0 + hi * 2 + 1][mn];
            acc = __builtin_amdgcn_wmma_f32_16x16x4_f32(
                /*neg_a=*/false, a, /*neg_b=*/false, b,
                /*c_mod=*/(short)0, acc, /*reuse_a=*/false, /*reuse_b=*/false);
        }
    }

    // C/D layout: vgpr v -> M = v + hi*8 ; N = mn. Fuse source-degree scaling.
#pragma unroll
    for (int v = 0; v < 8; ++v) {
        int m = rowBase + wave * 16 + hi * 8 + v;
        if (m < nRows) {
            float s = rowScale[m];
            Y[(size_t)m * FOUT + colBase + mn] = acc[v] * s;
        }
    }
}

// ---------------- SpMM scatter: AGG[dst[e], f] += H[src[e], f] ----------------

template<int FLOG>
__global__ void spmm_scatter(const int* __restrict__ src, const int* __restrict__ dst,
                             const float* __restrict__ H, float* __restrict__ AGG,
                             long long total)
{
    long long idx = (long long)blockIdx.x * blockDim.x + threadIdx.x;
    if (idx >= total) return;
    int e = (int)(idx >> FLOG);
    int f = (int)(idx & ((1 << FLOG) - 1));
    int s = src[e];
    int d = dst[e];
    float v = H[((size_t)s << FLOG) + f];
    atomicAdd(&AGG[((size_t)d << FLOG) + f], v);
}

// ---------------- fused dest-degree scale + bias (+ReLU) ----------------

template<int FLOG, bool RELU>
__global__ void scale_bias(float* __restrict__ A, const float* __restrict__ degIn,
                           const float* __restrict__ bias, long long total)
{
    long long idx = (long long)blockIdx.x * blockDim.x + threadIdx.x;
    if (idx >= total) return;
    int n = (int)(idx >> FLOG);
    int f = (int)(idx & ((1 << FLOG) - 1));
    float v = A[idx] * degIn[n] + bias[f];
    if (RELU) v = fmaxf(v, 0.0f);
    A[idx] = v;
}

// ---------------- launcher ----------------

static inline int cdiv_ll(long long a, long long b) { return (int)((a + b - 1) / b); }

extern "C" void kernel_launch(void* const* d_in, const int* in_sizes, int n_in,
                              void* d_out, int out_size, void* d_ws, size_t ws_size,
                              hipStream_t stream) {
    const float* x  = (const float*)d_in[0];
    const int*   src = (const int*)d_in[1];
    const int*   dst = (const int*)d_in[2];
    const float* W1 = (const float*)d_in[3];
    const float* b1 = (const float*)d_in[4];
    const float* W2 = (const float*)d_in[5];
    const float* b2 = (const float*)d_in[6];
    float* out = (float*)d_out;

    const int N = in_sizes[0] / 128;   // nodes
    const int E = in_sizes[1];         // edges

    float* ws     = (float*)d_ws;
    float* degOut = ws;                          // N floats (becomes deg_out^-1/2)
    float* degIn  = ws + (long long)N;           // N floats (becomes deg_in^-1/2)
    float* bufH   = ws + 2LL * N;                // N*128 (layer1 XW; reused for layer2 HW, N*64)
    float* bufAgg = ws + 2LL * N + 128LL * N;    // N*128 aggregation buffer

    // ---- degrees ----
    zero_f32<<<cdiv_ll(2LL * N, TPB), TPB, 0, stream>>>(degOut, 2LL * N);
    degree_count<<<cdiv_ll(E, TPB), TPB, 0, stream>>>(src, dst, degOut, degIn, E);
    degree_finalize<<<cdiv_ll(2LL * N, TPB), TPB, 0, stream>>>(degOut, 2LL * N);

    // ---- layer 1: h = relu( D_in^-1/2 * A^T * (D_out^-1/2 * x W1) + b1 ) ----
    dim3 g1(cdiv_ll(N, 128), 128 / 16);
    gemm_rowscale<128><<<g1, TPB, 0, stream>>>(x, W1, degOut, bufH, N);
    zero_f32<<<cdiv_ll(128LL * N, TPB), TPB, 0, stream>>>(bufAgg, 128LL * N);
    spmm_scatter<7><<<cdiv_ll((long long)E * 128, TPB), TPB, 0, stream>>>(
        src, dst, bufH, bufAgg, (long long)E * 128);
    scale_bias<7, true><<<cdiv_ll(128LL * N, TPB), TPB, 0, stream>>>(
        bufAgg, degIn, b1, 128LL * N);

    // ---- layer 2: out = D_in^-1/2 * A^T * (D_out^-1/2 * h W2) + b2 ----
    dim3 g2(cdiv_ll(N, 128), 64 / 16);
    gemm_rowscale<64><<<g2, TPB, 0, stream>>>(bufAgg, W2, degOut, bufH, N);
    zero_f32<<<cdiv_ll(64LL * N, TPB), TPB, 0, stream>>>(out, 64LL * N);
    spmm_scatter<6><<<cdiv_ll((long long)E * 64, TPB), TPB, 0, stream>>>(
        src, dst, bufH, out, (long long)E * 64);
    scale_bias<6, false><<<cdiv_ll(64LL * N, TPB), TPB, 0, stream>>>(
        out, degIn, b2, 64LL * N);
}